// PiecewiseLinear_64561948393782
// MI455X (gfx1250) — compile-verified
//
#include <hip/hip_runtime.h>

// ---------------------------------------------------------------------------
// Piecewise-linear interpolation, B rows x 256-bin monotone tables, 4096
// queries/row. Memory-bound (~136 MiB total traffic -> ~5.8us at 23.3 TB/s).
// CDNA5 paths exercised:
//   - global_load_async_to_lds_b64 + s_wait_asynccnt (ASYNCcnt) for the table
//   - native V_TANH_F32 transcendental (replaces ocml's branchy expansion)
//   - nontemporal b128 streaming (TH=NT) so the 128 MiB stream skips L2 reuse
//   - global_prefetch_b8 lookahead
// O(1) bin lookup (no binary search) exploiting x_abs[i] in [i*bw, (i+1)*bw];
// 2 ds_load_b64 per query (probe pair + other endpoint), rcp-based slope.
// ---------------------------------------------------------------------------

typedef float v4f __attribute__((ext_vector_type(4)));
typedef float v2f __attribute__((ext_vector_type(2)));

constexpr int   BINS = 256;
constexpr int   NQ   = 4096;
constexpr float BW   = 1.0f / (float)BINS;

// __shared__ pointers carry the LDS byte offset in their low 32 bits
// (aperture bits live in [63:32]); this is the VDST operand the async
// global->LDS instruction expects.
__device__ __forceinline__ unsigned lds_byte_offset(const void* p) {
  return (unsigned)(unsigned long long)p;
}

// Native CDNA5 tanh (V_TANH_F32) instead of the ocml exp-based expansion.
__device__ __forceinline__ float fast_tanh(float x) {
#if __has_builtin(__builtin_amdgcn_tanhf)
  return __builtin_amdgcn_tanhf(x);
#elif __has_builtin(__builtin_amdgcn_tanh_f32)
  return __builtin_amdgcn_tanh_f32(x);
#else
  float r;
  asm("v_tanh_f32 %0, %1" : "=v"(r) : "v"(x));
  return r;
#endif
}

// v_rcp_f32 (1-ulp) instead of the IEEE div_scale/div_fmas/div_fixup chain.
__device__ __forceinline__ float fast_rcp(float x) {
  return __builtin_amdgcn_rcpf(x);
}

__global__ __launch_bounds__(256, 4)
void pwl_interp_kernel(const float* __restrict__ params,  // [B, BINS, 2]
                       const float* __restrict__ xq,      // [B, NQ]
                       float* __restrict__ out)           // [B, NQ]
{
  __shared__ v2f tab[BINS];  // .x = x_abs (monotone), .y = t

  const int b   = blockIdx.x;
  const int tid = threadIdx.x;

  // ---- Phase 1: CDNA5 async copy of this lane's raw (p0, t) pair to LDS ----
  const v2f* prow = reinterpret_cast<const v2f*>(params) + (size_t)b * BINS;
  {
    unsigned   dst = lds_byte_offset(&tab[tid]);
    const v2f* src = prow + tid;
    asm volatile("global_load_async_to_lds_b64 %0, %1, off"
                 :: "v"(dst), "v"(src) : "memory");
#if __has_builtin(__builtin_amdgcn_s_wait_asynccnt)
    __builtin_amdgcn_s_wait_asynccnt(0);
#else
    asm volatile("s_wait_asynccnt 0" ::: "memory");
#endif
  }

  // Transform own slot in place: x_abs = clip(center + tanh(p0)*bw/2).
  // Each lane touches only its own slot -> no cross-lane hazard before the
  // barrier; own-wave async completion is guaranteed by s_wait_asynccnt above.
  {
    v2f   raw    = tab[tid];
    float center = ((float)tid + 0.5f) * BW;
    float xa     = fmaf(fast_tanh(raw.x), 0.5f * BW, center);
    xa           = fminf(fmaxf(xa, 0.0f), 1.0f);
    v2f cooked;  cooked.x = xa;  cooked.y = raw.y;
    tab[tid] = cooked;
  }
  __syncthreads();

  // ---- Phase 2: stream queries as float4, O(1) segment lookup in LDS ----
  const v4f* __restrict__ xrow = reinterpret_cast<const v4f*>(xq + (size_t)b * NQ);
  v4f*       __restrict__ orow = reinterpret_cast<v4f*>(out + (size_t)b * NQ);

  constexpr int ITERS = NQ / 4 / 256;  // = 4
#pragma unroll
  for (int it = 0; it < ITERS; ++it) {
    const int j = it * 256 + tid;
    if (it + 1 < ITERS) {
      __builtin_prefetch(&xrow[j + 256], 0, 0);  // global_prefetch_b8
    }
    v4f q = __builtin_nontemporal_load(&xrow[j]);  // b128, TH=NT (no reuse)
    v4f r;
#pragma unroll
    for (int c = 0; c < 4; ++c) {
      float xv = q[c];
      // Containing bin guess: x_abs[i] lies inside [i*bw, (i+1)*bw], so the
      // segment is (i-1,i) or (i,i+1) depending on xv vs x_abs[i].
      int i = (int)(xv * (float)BINS);
      i = i < 0 ? 0 : (i > BINS - 1 ? BINS - 1 : i);
      v2f probe = tab[i];                       // ds_load_b64 (xa_i, t_i)
      bool below = (xv < probe.x);
      int k2 = below ? i - 1 : i + 1;           // the other endpoint index
      // Edge fixups keep the extrapolation segments exact:
      //   i==0   && below  -> segment (0,1):     k2 = 1
      //   i==255 && !below -> segment (254,255): k2 = 254
      k2 = k2 < 0 ? 1 : (k2 > BINS - 1 ? BINS - 2 : k2);
      v2f other = tab[k2];                      // ds_load_b64
      bool swap = (k2 < i);                     // other is the left endpoint
      v2f p0 = swap ? other : probe;
      v2f p1 = swap ? probe : other;
      float slope = (p1.y - p0.y) * fast_rcp(p1.x - p0.x);
      r[c] = fmaf(slope, xv - p0.x, p0.y);
    }
    __builtin_nontemporal_store(r, &orow[j]);  // b128, TH=NT (write-once)
  }
}

extern "C" void kernel_launch(void* const* d_in, const int* in_sizes, int n_in,
                              void* d_out, int out_size, void* d_ws, size_t ws_size,
                              hipStream_t stream) {
  const float* params = (const float*)d_in[0];  // [B, 256, 2] f32
  const float* x      = (const float*)d_in[1];  // [B, 4096]   f32
  float*       out    = (float*)d_out;          // [B, 4096]   f32

  const int B = in_sizes[0] / (BINS * 2);       // 4096
  pwl_interp_kernel<<<dim3(B), dim3(256), 0, stream>>>(params, x, out);

  (void)n_in; (void)out_size; (void)d_ws; (void)ws_size;
}